// RegionLoss_24790551232876
// MI455X (gfx1250) — compile-verified
//
#include <hip/hip_runtime.h>

#define NB 32
#define NH 38
#define NW 38
#define NC 80
#define NA 5
#define MAXGT 50
#define CH_STRIDE (NH * NW)                 // 1444
#define CELLS_PER_B (NA * NH * NW)          // 7220
#define CHUNKS ((CELLS_PER_B + 255) / 256)  // 29
#define NPART (NB * CHUNKS)                 // 928
#define RECSZ 12                            // floats per GT record
#define NRECF (MAXGT * RECSZ)               // 600 floats per batch
#define OBJ_SCALE_F 5.0f
#define SIL_THRESH_F 0.6f

typedef float v2f __attribute__((ext_vector_type(2)));
typedef float v8f __attribute__((ext_vector_type(8)));

__constant__ float c_aw[NA] = {1.3221f, 3.19275f, 5.05587f, 9.47112f, 11.2364f};
__constant__ float c_ah[NA] = {1.73145f, 4.00944f, 8.09892f, 4.84053f, 10.0071f};

__device__ __forceinline__ float sigmoidf_(float v) {
    return 1.0f / (1.0f + expf(-v));
}

__device__ __forceinline__ float iou_box(float x1, float y1, float w1, float h1,
                                         float x2, float y2, float w2, float h2) {
    float mx = fminf(x1 - w1 * 0.5f, x2 - w2 * 0.5f);
    float Mx = fmaxf(x1 + w1 * 0.5f, x2 + w2 * 0.5f);
    float my = fminf(y1 - h1 * 0.5f, y2 - h2 * 0.5f);
    float My = fmaxf(y1 + h1 * 0.5f, y2 + h2 * 0.5f);
    float cw = w1 + w2 - (Mx - mx);
    float ch = h1 + h2 - (My - my);
    float inter = (cw <= 0.f || ch <= 0.f) ? 0.f : cw * ch;
    return inter / (w1 * h1 + w2 * h2 - inter);
}

// Workspace layout (floats):
//   [0 .. NB*600)           GT records: per (b,t) 12 floats:
//     {gx, gy, gw, gh, valid, key(-1 if invalid), tx, ty, tw, th, tconf, cls}
//     key = a*1444 + j*38 + i == linear cell index within the batch image.
//   [NB*600 .. NB*600+928)  per-block partial sums

// One thread per batch image; serial over t. Ownership (last-write-wins) is
// resolved later by forward-scanning records in the main kernel, which matches
// the reference's sequential scatter exactly.
__global__ void rl_gt(const float* __restrict__ out, const float* __restrict__ tgt,
                      float* __restrict__ ws) {
    int b = threadIdx.x;
    if (b >= NB) return;
    const float* tb = tgt + (size_t)b * MAXGT * 5;
    float* recs = ws + (size_t)b * NRECF;
    int validrun = 1;
    for (int t = 0; t < MAXGT; ++t) {
        float cls = tb[t * 5 + 0];
        float cx = tb[t * 5 + 1], cy = tb[t * 5 + 2];
        float cw = tb[t * 5 + 3], chh = tb[t * 5 + 4];
        if (cx == 0.f) validrun = 0;  // cumprod semantics
        float gx = cx * NW, gy = cy * NH, gw = cw * NW, gh = chh * NH;
        float* r = recs + t * RECSZ;
        r[0] = gx; r[1] = gy; r[2] = gw; r[3] = gh;
        r[4] = validrun ? 1.f : 0.f;
        if (!validrun) {
            r[5] = -1.f;  // never matches any cell
            r[6] = r[7] = r[8] = r[9] = r[10] = r[11] = 0.f;
            continue;
        }
        // best anchor: centered IoU reduces to min-overlap; first max wins (>)
        int best = 0; float bi = -1.f;
        for (int a = 0; a < NA; ++a) {
            float inter = fminf(c_aw[a], gw) * fminf(c_ah[a], gh);
            float io = inter / (c_aw[a] * c_ah[a] + gw * gh - inter);
            if (io > bi) { bi = io; best = a; }
        }
        int gi = (int)gx, gj = (int)gy;  // in-range: coords in [0.02,0.98]

        size_t cb = (((size_t)b * (NA * (5 + NC)) + best * (5 + NC)) * NH + gj) * NW + gi;
        float o0 = out[cb];
        float o1 = out[cb + 1 * (size_t)CH_STRIDE];
        float o2 = out[cb + 2 * (size_t)CH_STRIDE];
        float o3 = out[cb + 3 * (size_t)CH_STRIDE];
        float px = sigmoidf_(o0) + (float)gi;
        float py = sigmoidf_(o1) + (float)gj;
        float pw = expf(o2) * c_aw[best];
        float ph = expf(o3) * c_ah[best];

        r[5]  = (float)((best * NH + gj) * NW + gi);   // cell key (exact int in fp32)
        r[6]  = gx - (float)gi;                        // tx
        r[7]  = gy - (float)gj;                        // ty
        r[8]  = logf(gw / c_aw[best]);                 // tw
        r[9]  = logf(gh / c_ah[best]);                 // th
        r[10] = iou_box(gx, gy, gw, gh, px, py, pw, ph);  // tconf
        r[11] = cls;                                   // tcls
    }
}

// Main loss kernel: grid (NB, CHUNKS) x 256. Channels 0..4 read for all cells;
// class logits only at GT cells (<=1600 total). GT records staged into LDS via
// the CDNA5 async-to-LDS path; max-IoU and scatter-ownership resolved in one
// 50-iteration loop per cell.
__global__ void rl_main(const float* __restrict__ out, const float* __restrict__ wsc,
                        float* __restrict__ partials) {
    __shared__ float sGT[NRECF + 2];
    __shared__ float sred[256];
    const int b = blockIdx.x;
    const int tid = threadIdx.x;

    const float* recs = wsc + (size_t)b * NRECF;
    for (int idx = tid; idx < NRECF; idx += 256) {
        unsigned lds = (unsigned)(unsigned long long)(size_t)&sGT[idx];
        unsigned long long ga = (unsigned long long)(size_t)(recs + idx);
        asm volatile("global_load_async_to_lds_b32 %0, %1, off"
                     :: "v"(lds), "v"(ga) : "memory");
    }
    asm volatile("s_wait_asynccnt 0" ::: "memory");
    __syncthreads();

    int c = blockIdx.y * 256 + tid;
    float acc = 0.f;
    if (c < CELLS_PER_B) {
        int a = c / CH_STRIDE;
        int rem = c - a * CH_STRIDE;
        int j = rem / NW;
        int i = rem - j * NW;
        size_t base = (((size_t)b * (NA * (5 + NC)) + a * (5 + NC)) * NH + j) * NW + i;
        __builtin_prefetch(&out[base], 0, 1);  // global_prefetch_b8
        float o0 = out[base];
        float o1 = out[base + 1 * (size_t)CH_STRIDE];
        float o2 = out[base + 2 * (size_t)CH_STRIDE];
        float o3 = out[base + 3 * (size_t)CH_STRIDE];
        float o4 = out[base + 4 * (size_t)CH_STRIDE];
        float x = sigmoidf_(o0), y = sigmoidf_(o1), conf = sigmoidf_(o4);
        float px = x + (float)i, py = y + (float)j;
        float pw = expf(o2) * c_aw[a], ph = expf(o3) * c_ah[a];

        float px1 = px - pw * 0.5f, px2 = px + pw * 0.5f;
        float py1 = py - ph * 0.5f, py2 = py + ph * 0.5f;
        float areap = pw * ph;
        float ckey = (float)c;

        // targets default to base values; overwritten by matching GT records
        // in forward order => exact last-write-wins scatter semantics.
        float txv = 0.5f, tyv = 0.5f, twv = 0.f, thv = 0.f;
        float tcf = 0.f, tcl = 0.f, ig = 0.f;
        float mi = 0.f;
        #pragma unroll 2
        for (int t = 0; t < MAXGT; ++t) {
            const float* r = &sGT[t * RECSZ];
            float gx = r[0], gy = r[1], gw = r[2], gh = r[3], vf = r[4];
            float mx = fminf(px1, gx - gw * 0.5f);
            float Mx = fmaxf(px2, gx + gw * 0.5f);
            float my = fminf(py1, gy - gh * 0.5f);
            float My = fmaxf(py2, gy + gh * 0.5f);
            float cw = pw + gw - (Mx - mx);
            float chh = ph + gh - (My - my);
            float inter = (cw <= 0.f || chh <= 0.f) ? 0.f : cw * chh;
            float io = vf * (inter / (areap + gw * gh - inter));
            mi = fmaxf(mi, io);
            if (r[5] == ckey) {  // this GT owns the cell (later t overwrites)
                txv = r[6]; tyv = r[7]; twv = r[8]; thv = r[9];
                tcf = r[10]; tcl = r[11]; ig = 1.f;
            }
        }

        float cmask = (ig > 0.5f) ? OBJ_SCALE_F : ((mi > SIL_THRESH_F) ? 0.f : 1.f);
        float dx = x - txv, dy = y - tyv, dw = o2 - twv, dh = o3 - thv, dc = conf - tcf;
        acc = 0.5f * (dx * dx + dy * dy + dw * dw + dh * dh + cmask * dc * dc);

        if (ig > 0.5f) {  // class CE only at GT cells
            size_t cb = base + 5 * (size_t)CH_STRIDE;
            float m = out[cb];
            for (int k = 1; k < NC; ++k) m = fmaxf(m, out[cb + (size_t)k * CH_STRIDE]);
            float s = 0.f;
            for (int k = 0; k < NC; ++k) s += expf(out[cb + (size_t)k * CH_STRIDE] - m);
            int lbl = (int)tcl;
            acc += (m + logf(s)) - out[cb + (size_t)lbl * CH_STRIDE];
        }
    }

    sred[tid] = acc;
    __syncthreads();
    for (int s = 128; s > 0; s >>= 1) {
        if (tid < s) sred[tid] += sred[tid + s];
        __syncthreads();
    }
    if (tid == 0) partials[blockIdx.x * gridDim.y + blockIdx.y] = sred[0];
}

// Single-wave final reduction: 32 lane partials summed with one
// V_WMMA_F32_16X16X4_F32 against an all-ones B (D[m][n] = s_m + s_{m+16}).
__global__ void rl_final(const float* __restrict__ partials, float* __restrict__ outp) {
    __shared__ float sh[32];
    int l = threadIdx.x;
    float s = 0.f;
    for (int i = l; i < NPART; i += 32) s += partials[i];
    v2f a;  a[0] = s;    a[1] = 0.f;   // A 16x4: row m gets K0 = s_m, K2 = s_{m+16}
    v2f bo; bo[0] = 1.f; bo[1] = 1.f;  // B 4x16: all ones (layout-agnostic)
    v8f cz = {0.f, 0.f, 0.f, 0.f, 0.f, 0.f, 0.f, 0.f};
    v8f d = __builtin_amdgcn_wmma_f32_16x16x4_f32(
        false, a, false, bo, (short)0, cz, false, false);
    float p = d[0] + d[1] + d[2] + d[3] + d[4] + d[5] + d[6] + d[7];
    sh[l] = p;
    __syncthreads();
    if (l == 0) outp[0] = sh[0] + sh[16];  // rows 0..7 + rows 8..15
}

extern "C" void kernel_launch(void* const* d_in, const int* in_sizes, int n_in,
                              void* d_out, int out_size, void* d_ws, size_t ws_size,
                              hipStream_t stream) {
    (void)in_sizes; (void)n_in; (void)out_size; (void)ws_size;
    const float* output = (const float*)d_in[0];
    const float* target = (const float*)d_in[1];
    float* ws = (float*)d_ws;
    float* partials = ws + (size_t)NB * NRECF;

    rl_gt<<<1, NB, 0, stream>>>(output, target, ws);
    rl_main<<<dim3(NB, CHUNKS), 256, 0, stream>>>(output, ws, partials);
    rl_final<<<1, 32, 0, stream>>>(partials, (float*)d_out);
}